// RTDETRPostProcessor_19198503813628
// MI455X (gfx1250) — compile-verified
//
#include <hip/hip_runtime.h>
#include <stdint.h>

// Problem constants (match reference)
#define BATCH 1024
#define NQ    300
#define NC    80
#define NELEM (NQ * NC)      // 24000 scores per batch image
#define TOPK  300
#define SORTN 512            // next pow2 >= TOPK
#define NTHREADS 256         // 8 wave32s

// LDS layout (dynamic shared):
//   float  logits[NELEM]      96000 B   (whole batch slab -- only possible on CDNA5's 320KB LDS)
//   uint   hist[256]           1024 B
//   uint2  buf[SORTN]          4096 B
//   uint   scal[8]               32 B
#define LOGITS_BYTES (NELEM * 4)
#define HIST_BYTES   (256 * 4)
#define BUF_BYTES    (SORTN * 8)
#define SMEM_BYTES   (LOGITS_BYTES + HIST_BYTES + BUF_BYTES + 32)

typedef unsigned int v4u __attribute__((ext_vector_type(4)));
typedef int          v8i __attribute__((ext_vector_type(8)));
typedef int          v4i __attribute__((ext_vector_type(4)));

// Order-preserving float -> uint key (monotonic: bigger float => bigger key)
__device__ __forceinline__ unsigned orderkey(float f) {
    unsigned u = __float_as_uint(f);
    return u ^ ((u & 0x80000000u) ? 0xFFFFFFFFu : 0x80000000u);
}

__global__ void rtdetr_post_kernel(const float* __restrict__ pred_logits,  // [B,Q,C]
                                   const float* __restrict__ pred_boxes,   // [B,Q,4]
                                   const float* __restrict__ orig_sizes,   // [B,2]
                                   float* __restrict__ out)                // labels|boxes|scores
{
    extern __shared__ unsigned char smem_raw[];
    float*    s_logits = reinterpret_cast<float*>(smem_raw);
    unsigned* s_hist   = reinterpret_cast<unsigned*>(smem_raw + LOGITS_BYTES);
    uint2*    s_buf    = reinterpret_cast<uint2*>(smem_raw + LOGITS_BYTES + HIST_BYTES);
    unsigned* s_scal   = reinterpret_cast<unsigned*>(smem_raw + LOGITS_BYTES + HIST_BYTES + BUF_BYTES);

    const int tid = threadIdx.x;
    const int b   = blockIdx.x;

    // ---------------------------------------------------------------
    // Stage 1: TDM DMA of this batch's 96KB score slab HBM -> LDS.
    // One descriptor, issued by one wave; VALU/VMEM pipes stay free.
    // D# layout per CDNA5 ISA ch.8 (group0 128b, group1 256b).
    // ---------------------------------------------------------------
    if (tid == 0) {
        const unsigned lds_addr = (unsigned)(uintptr_t)(void*)s_logits;      // LDS byte offset
        const unsigned long long ga =
            (unsigned long long)(uintptr_t)(pred_logits + (size_t)b * NELEM);

        v4u g0;
        g0.x = 1u;                                            // count=1 (valid user D#)
        g0.y = lds_addr;                                      // lds_addr
        g0.z = (unsigned)ga;                                  // global_addr[31:0]
        g0.w = (unsigned)((ga >> 32) & 0x1FFFFFFull)          // global_addr[56:32]
             | (2u << 30);                                    // type=2 ("image")

        v8i g1;
        g1[0] = (int)(2u << 16);                              // data_size=2 -> 4 bytes
        g1[1] = (int)(((unsigned)NELEM & 0xFFFFu) << 16);     // tensor_dim0[15:0]
        g1[2] = (int)((((unsigned)NELEM >> 16) & 0xFFFFu)     // tensor_dim0[31:16]
             | (1u << 16));                                   // tensor_dim1 = 1
        g1[3] = (int)(((unsigned)NELEM & 0xFFFFu) << 16);     // tile_dim0 = 24000
        g1[4] = 1;                                            // tile_dim1 = 1, tile_dim2 = 0
        g1[5] = NELEM;                                        // tensor_dim0_stride lo
        g1[6] = 0;                                            // stride hi / dim1_stride lo
        g1[7] = 0;

        v4i g2 = {0, 0, 0, 0};                                // group2 (unused, <=2D tensor)
        v4i g3 = {0, 0, 0, 0};                                // group3 (unused)
        v8i g4 = {0, 0, 0, 0, 0, 0, 0, 0};                    // extra group (clang-23 form)

        __builtin_amdgcn_tensor_load_to_lds(g0, g1, g2, g3, g4, 0);
        __builtin_amdgcn_s_wait_tensorcnt(0);                 // s_wait_tensorcnt 0
    }
    __syncthreads();

    // ---------------------------------------------------------------
    // Stage 2: radix select (4 x 8-bit MSD rounds) entirely from LDS.
    // Finds exact key threshold T of the K-th largest logit.
    // ---------------------------------------------------------------
    if (tid == 0) { s_scal[0] = 0u; s_scal[1] = TOPK; }       // prefix, remaining
    __syncthreads();

    for (int pass = 0; pass < 4; ++pass) {
        s_hist[tid] = 0u;                                     // 256 threads, 256 bins
        __syncthreads();

        const unsigned prefix = s_scal[0];
        const int shift = 24 - pass * 8;
        for (int i = tid; i < NELEM; i += NTHREADS) {
            unsigned key = orderkey(s_logits[i]);
            bool match = (pass == 0) || ((key >> (shift + 8)) == prefix);
            if (match) atomicAdd(&s_hist[(key >> shift) & 0xFFu], 1u);
        }
        __syncthreads();

        if (tid == 0) {
            unsigned remaining = s_scal[1];
            unsigned cum = 0;
            int v = 255;
            for (; v > 0; --v) {
                cum += s_hist[v];
                if (cum >= remaining) break;
            }
            if (cum < remaining) cum += s_hist[0];            // v==0 fallthrough
            s_scal[1] = remaining - (cum - s_hist[v]);        // still needed inside bin v
            s_scal[0] = (prefix << 8) | (unsigned)v;
        }
        __syncthreads();
    }

    const unsigned T        = s_scal[0];                      // key of K-th largest
    const unsigned need_eq  = s_scal[1];                      // #elems == T to keep
    const unsigned cnt_gt   = TOPK - need_eq;                 // #elems  > T

    // ---------------------------------------------------------------
    // Stage 3: compact survivors into s_buf, pad, bitonic sort
    // (desc by key, asc by index on ties).
    // ---------------------------------------------------------------
    if (tid == 0) { s_scal[2] = 0u; s_scal[3] = 0u; }
    __syncthreads();

    for (int i = tid; i < NELEM; i += NTHREADS) {
        unsigned key = orderkey(s_logits[i]);
        if (key > T) {
            unsigned p = atomicAdd(&s_scal[2], 1u);
            s_buf[p] = make_uint2(key, (unsigned)i);
        } else if (key == T) {
            unsigned p = atomicAdd(&s_scal[3], 1u);
            if (p < need_eq) s_buf[cnt_gt + p] = make_uint2(key, (unsigned)i);
        }
    }
    for (int i = TOPK + tid; i < SORTN; i += NTHREADS)
        s_buf[i] = make_uint2(0u, 0xFFFFFFFFu);               // sentinels sort last
    __syncthreads();

    for (unsigned k = 2; k <= SORTN; k <<= 1) {
        for (unsigned j = k >> 1; j > 0; j >>= 1) {
            for (unsigned i = tid; i < SORTN; i += NTHREADS) {
                unsigned ixj = i ^ j;
                if (ixj > i) {
                    uint2 a = s_buf[i];
                    uint2 c = s_buf[ixj];
                    bool up = ((i & k) == 0);                 // comparator-order block
                    bool aFirst = (a.x > c.x) || (a.x == c.x && a.y < c.y);
                    if (up ? !aFirst : aFirst) { s_buf[i] = c; s_buf[ixj] = a; }
                }
            }
            __syncthreads();
        }
    }

    // ---------------------------------------------------------------
    // Stage 4: epilogue — sigmoid on the 300 winners, decode, box
    // gather + cxcywh->xyxy + scale, coalesced stores.
    // d_out layout: labels[B*K] | boxes[B*K*4] | scores[B*K]
    // ---------------------------------------------------------------
    const float ow = orig_sizes[2 * b + 0];
    const float oh = orig_sizes[2 * b + 1];
    float* out_labels = out;
    float* out_boxes  = out + (size_t)BATCH * TOPK;
    float* out_scores = out + (size_t)BATCH * TOPK * 5;
    const float4* boxes_in = reinterpret_cast<const float4*>(pred_boxes);

    for (int i = tid; i < TOPK; i += NTHREADS) {
        uint2 e = s_buf[i];
        unsigned idx = e.y;
        float logit = s_logits[idx];
        float score = 1.0f / (1.0f + __expf(-logit));
        unsigned label = idx % NC;
        unsigned q     = idx / NC;

        float4 bx = boxes_in[(size_t)b * NQ + q];             // cx, cy, w, h
        float x0 = (bx.x - 0.5f * bx.z) * ow;
        float y0 = (bx.y - 0.5f * bx.w) * oh;
        float x1 = (bx.x + 0.5f * bx.z) * ow;
        float y1 = (bx.y + 0.5f * bx.w) * oh;

        size_t o = (size_t)b * TOPK + i;
        out_labels[o] = (float)label;
        float* ob = out_boxes + o * 4;
        ob[0] = x0; ob[1] = y0; ob[2] = x1; ob[3] = y1;
        out_scores[o] = score;
    }
}

extern "C" void kernel_launch(void* const* d_in, const int* in_sizes, int n_in,
                              void* d_out, int out_size, void* d_ws, size_t ws_size,
                              hipStream_t stream) {
    const float* pred_logits = (const float*)d_in[0];  // [B,Q,C] f32
    const float* pred_boxes  = (const float*)d_in[1];  // [B,Q,4] f32
    const float* orig_sizes  = (const float*)d_in[2];  // [B,2]   f32
    float* out = (float*)d_out;

    (void)in_sizes; (void)n_in; (void)out_size; (void)d_ws; (void)ws_size;

    rtdetr_post_kernel<<<dim3(BATCH), dim3(NTHREADS), SMEM_BYTES, stream>>>(
        pred_logits, pred_boxes, orig_sizes, out);
}